// ContainmentLoss_40862318854276
// MI455X (gfx1250) — compile-verified
//
#include <hip/hip_runtime.h>
#include <stdint.h>

// ContainmentLoss for MI455X (gfx1250).
// Design: one workgroup per batch image; the whole 256KB boundary image lives
// in the WGP's 320KB LDS (a CDNA5-only capability); 256 sequential distance-
// transform steps are synchronized with workgroup barriers instead of kernel
// launches. Raw input staged with async global->LDS (ASYNCcnt) copies; final
// 1024-way reduction done on the matrix unit (v_wmma_f32_16x16x4_f32, B=ones).

constexpr int   Himg   = 256;
constexpr int   Wimg   = 256;
constexpr int   NPIX   = Himg * Wimg;      // 65536 px = 256 KB f32
constexpr int   TPB    = 1024;             // 32 waves (wave32) per workgroup
constexpr int   PPT    = 64;               // contiguous pixels per thread (1/4 row)
constexpr int   NITERS = 256;              // ceil(256 / (DT_K//2))
constexpr float DT_HC  = 0.35f;

typedef __attribute__((ext_vector_type(2))) float v2f;
typedef __attribute__((ext_vector_type(8))) float v8f;

// Async-copy 64 contiguous floats (16 x B128) from global into LDS at byte
// offset pix0*4, then drain ASYNCcnt for this wave.
__device__ __forceinline__ void async_stage_row(const float* gsrc, int pix0) {
  unsigned long long gaddr = (unsigned long long)(uintptr_t)gsrc;
  unsigned int       laddr = (unsigned int)(pix0 * 4);
#pragma unroll
  for (int k = 0; k < 16; ++k) {
    asm volatile("global_load_async_to_lds_b128 %0, %1, off"
                 :: "v"(laddr + 16u * (unsigned)k),
                    "v"(gaddr + 16ull * (unsigned long long)k)
                 : "memory");
  }
  asm volatile("s_wait_asynccnt 0x0" ::: "memory");
}

__global__ void __launch_bounds__(TPB, 1)
containment_main(const float* __restrict__ pred,
                 const float* __restrict__ target,
                 float* __restrict__ partials)
{
  extern __shared__ float buf[];                 // 65536 floats (dynamic LDS)
  const int b    = blockIdx.x;
  const int tid  = threadIdx.x;
  const int row  = tid >> 2;
  const int col0 = (tid & 3) * PPT;
  const int pix0 = row * Wimg + col0;

  const float* tgt0 = target + (size_t)b * 5 * NPIX + pix0;          // ch 0
  const float* fbl0 = pred + ((size_t)b * 5 + 1) * NPIX + pix0;      // ch 1

  // 3x3 DT kernel weights: exp(-hypot/0.35); center weight is 1.
  const float ke = expf(-1.0f / DT_HC);              // edge  (dist 1)
  const float kc = expf(-1.41421356237f / DT_HC);    // corner (dist sqrt2)

  float outacc[PPT];

  // ===== stage target ch0 into LDS (async), sigmoid in place =====
  async_stage_row(tgt0, pix0);
  __syncthreads();
#pragma unroll 8
  for (int x = 0; x < PPT; ++x) {
    float v = buf[pix0 + x];
    buf[pix0 + x] = 1.0f / (1.0f + expf(-10.0f * (v - 0.5f)));
  }
  __syncthreads();

  // ===== 5x5 dilation (rolling column-max) -> outside, staged in outacc =====
  {
    const int r0 = max(row - 2, 0), r1 = max(row - 1, 0);
    const int r3 = min(row + 1, Himg - 1), r4 = min(row + 2, Himg - 1);
    auto colmax = [&](int c) {
      c = min(max(c, 0), Wimg - 1);
      float m = buf[r0 * Wimg + c];
      m = fmaxf(m, buf[r1 * Wimg + c]);
      m = fmaxf(m, buf[row * Wimg + c]);
      m = fmaxf(m, buf[r3 * Wimg + c]);
      m = fmaxf(m, buf[r4 * Wimg + c]);
      return m;
    };
    float m0 = colmax(col0 - 2), m1 = colmax(col0 - 1);
    float m2 = colmax(col0),     m3 = colmax(col0 + 1);
#pragma unroll
    for (int x = 0; x < PPT; ++x) {
      float m4  = colmax(col0 + x + 2);
      float dil = fmaxf(fmaxf(fmaxf(m0, m1), fmaxf(m2, m3)), m4);
      outacc[x] = 1.0f - dil;                        // outside = initial boundary
      m0 = m1; m1 = m2; m2 = m3; m3 = m4;
    }
  }
  __syncthreads();                                   // all mask reads done
#pragma unroll
  for (int x = 0; x < PPT; ++x) buf[pix0 + x] = outacc[x];   // boundary init
#pragma unroll
  for (int x = 0; x < PPT; ++x) outacc[x] = 0.0f;            // dist accumulator
  __syncthreads();

  // ===== 256 sequential DT steps, boundary fully LDS-resident =====
  const int rm = max(row - 1, 0) * Wimg;
  const int rz = row * Wimg;
  const int rp = min(row + 1, Himg - 1) * Wimg;
  const int cl = max(col0 - 1, 0);

  for (int it = 0; it < NITERS; ++it) {
    const float offset = (float)((3 * it) >> 1);     // (i*DT_K)//2 per reference
    unsigned int mlo = 0u, mhi = 0u;
    // rolling 3x3 window: 3 new LDS loads per output pixel
    float a0 = buf[rm + cl], a1 = buf[rm + col0];
    float b0 = buf[rz + cl], b1 = buf[rz + col0];
    float c0 = buf[rp + cl], c1 = buf[rp + col0];
#pragma unroll
    for (int x = 0; x < PPT; ++x) {
      const int cn = min(col0 + x + 1, Wimg - 1);
      float a2 = buf[rm + cn];
      float b2 = buf[rz + cn];
      float c2 = buf[rp + cn];
      float conv = kc * (a0 + a2 + c0 + c2) + ke * (a1 + c1 + b0 + b2) + b1;
      // cdt = -h*log(conv); +inf (conv==0) -> 0; mask = cdt > 0  <=>  0<conv<1
      if (conv > 0.0f && conv < 1.0f) {
        outacc[x] += offset - DT_HC * logf(conv);
        if (x < 32) mlo |= 1u << x; else mhi |= 1u << (x - 32);
      }
      a0 = a1; a1 = a2; b0 = b1; b1 = b2; c0 = c1; c1 = c2;
    }
    __syncthreads();                 // everyone done reading old boundary
    if (mlo | mhi) {
#pragma unroll
      for (int x = 0; x < 32; ++x)
        if ((mlo >> x) & 1u) buf[pix0 + x] = 1.0f;
#pragma unroll
      for (int x = 0; x < 32; ++x)
        if ((mhi >> x) & 1u) buf[pix0 + 32 + x] = 1.0f;
    }
    __syncthreads();                 // boundary updated for next step
  }

  // ===== re-stage mask (boundary destroyed it), final violation sum =====
  async_stage_row(tgt0, pix0);
  __syncthreads();
#pragma unroll 8
  for (int x = 0; x < PPT; ++x) {
    float v = buf[pix0 + x];
    buf[pix0 + x] = 1.0f / (1.0f + expf(-10.0f * (v - 0.5f)));
  }
  __syncthreads();

  float vsum = 0.0f;
  {
    const int r0 = max(row - 2, 0), r1 = max(row - 1, 0);
    const int r3 = min(row + 1, Himg - 1), r4 = min(row + 2, Himg - 1);
    auto colmax = [&](int c) {
      c = min(max(c, 0), Wimg - 1);
      float m = buf[r0 * Wimg + c];
      m = fmaxf(m, buf[r1 * Wimg + c]);
      m = fmaxf(m, buf[row * Wimg + c]);
      m = fmaxf(m, buf[r3 * Wimg + c]);
      m = fmaxf(m, buf[r4 * Wimg + c]);
      return m;
    };
    float m0 = colmax(col0 - 2), m1 = colmax(col0 - 1);
    float m2 = colmax(col0),     m3 = colmax(col0 + 1);
#pragma unroll
    for (int x = 0; x < PPT; ++x) {
      float m4      = colmax(col0 + x + 2);
      float dil     = fmaxf(fmaxf(fmaxf(m0, m1), fmaxf(m2, m3)), m4);
      float outside = 1.0f - dil;
      float pen     = fminf(outacc[x] * 0.1f, 1.0f);   // dist >= 0 always
      vsum += fbl0[x] * outside * pen;
      m0 = m1; m1 = m2; m2 = m3; m3 = m4;
    }
  }
  __syncthreads();                                   // mask reads done
  buf[tid] = vsum;                                   // 1024 partials in LDS
  __syncthreads();

  // ===== matrix-unit reduction: 16 accumulating WMMAs (B = ones) =====
  if (tid < 32) {                                    // wave 0, EXEC all-ones
    v8f acc = {0.f, 0.f, 0.f, 0.f, 0.f, 0.f, 0.f, 0.f};
    v2f ones; ones[0] = 1.0f; ones[1] = 1.0f;        // 4x16 ones B-matrix
#pragma unroll
    for (int g = 0; g < 16; ++g) {
      v2f a;                                         // 16x4 A tile = 64 partials
      a[0] = buf[g * 64 + tid * 2 + 0];
      a[1] = buf[g * 64 + tid * 2 + 1];
      acc = __builtin_amdgcn_wmma_f32_16x16x4_f32(false, a, false, ones,
                                                  (short)0, acc, false, false);
    }
    // D columns identical (B=1): lane0 holds rows M=0..7, lane16 rows M=8..15
    float s = acc[0] + acc[1] + acc[2] + acc[3] +
              acc[4] + acc[5] + acc[6] + acc[7];
    s += __shfl_xor(s, 16, 32);
    if (tid == 0) partials[b] = s;
  }
}

__global__ void containment_combine(const float* __restrict__ partials,
                                    float* __restrict__ out) {
  if (threadIdx.x == 0)
    out[0] = (partials[0] + partials[1] + partials[2] + partials[3]) *
             (1.0f / 262144.0f);                     // mean over 4*256*256
}

extern "C" void kernel_launch(void* const* d_in, const int* in_sizes, int n_in,
                              void* d_out, int out_size, void* d_ws, size_t ws_size,
                              hipStream_t stream) {
  const float* pred   = (const float*)d_in[0];
  const float* target = (const float*)d_in[1];
  float* partials = (float*)d_ws;                    // 4 floats of scratch
  float* out      = (float*)d_out;

  containment_main<<<4, TPB, NPIX * sizeof(float), stream>>>(pred, target, partials);
  containment_combine<<<1, 32, 0, stream>>>(partials, out);
}